// MoonVitPatchPacker_75084618268853
// MI455X (gfx1250) — compile-verified
//
#include <hip/hip_runtime.h>
#include <stdint.h>

// ---------------- problem constants (from reference) ----------------
#define B_    32
#define T_    4
#define C_    3
#define H_    448
#define W_    448
#define PS_   16
#define NR_   28                    // H/PS
#define NC_   28                    // W/PS
#define PPF_  (NR_*NC_)             // 784 patches per frame
#define TP_   (T_*PPF_)             // 3136 patches per batch (< MAX_PATCHES)
#define FEAT_ (C_*PS_*PS_)          // 768
#define HW_   (H_*W_)               // 200704
#define BTP_  (B_*TP_)              // 100352
#define PATCH_ELEMS ((size_t)BTP_*(size_t)FEAT_)   // 77,070,336 floats

// TDM task granularity: 7 patches (quarter patch-row) per descriptor.
#define QP_   7
#define QC_   (NC_/QP_)             // 4 tasks per (b,t,r) row
#define NTASK_ (B_*T_*NR_*QC_)      // 14336 tasks
#define TILE_FLOATS_ (QP_*FEAT_)    // 5376 floats = 21504 B per TDM transfer

typedef unsigned int v4u __attribute__((ext_vector_type(4)));
typedef int          v8i __attribute__((ext_vector_type(8)));
typedef int          v4i __attribute__((ext_vector_type(4)));

#if defined(__HIP_DEVICE_COMPILE__) && __has_builtin(__builtin_amdgcn_tensor_load_to_lds) && __has_builtin(__builtin_amdgcn_tensor_store_from_lds)
#define HAVE_TDM 1
#else
#define HAVE_TDM 0
#endif

#if HAVE_TDM
__device__ __forceinline__ void tdm_wait0() {
#if __has_builtin(__builtin_amdgcn_s_wait_tensorcnt)
  __builtin_amdgcn_s_wait_tensorcnt(0);
#else
  asm volatile("s_wait_tensorcnt 0x0" ::: "memory");
#endif
}

__device__ __forceinline__ void tdm_load(v4u g0, v8i g1, v4i g2, v4i g3) {
#if __clang_major__ >= 23
  v8i z8 = {};
  __builtin_amdgcn_tensor_load_to_lds(g0, g1, g2, g3, z8, 0);
#else
  __builtin_amdgcn_tensor_load_to_lds(g0, g1, g2, g3, 0);
#endif
}

__device__ __forceinline__ void tdm_store(v4u g0, v8i g1, v4i g2, v4i g3) {
#if __clang_major__ >= 23
  v8i z8 = {};
  __builtin_amdgcn_tensor_store_from_lds(g0, g1, g2, g3, z8, 0);
#else
  __builtin_amdgcn_tensor_store_from_lds(g0, g1, g2, g3, 0);
#endif
}
#endif // HAVE_TDM

// One wave (32 threads) per workgroup. Each task = 7 patches:
//   TDM load  : 3D tile x=16 (contig) / y=16 (stride 448) / ch=3 (stride 200704),
//               iterated 7x over patch columns (global += 16 elems, LDS += 768 elems)
//   TDM store : 1D contiguous 5376 floats -> output (patches are output-contiguous)
// Double-buffered LDS so store(i) overlaps load(i+1) (TENSORcnt is in-order per wave).
__global__ __launch_bounds__(32) void moonvit_patch_pack(const float* __restrict__ in,
                                                         float* __restrict__ out) {
#if HAVE_TDM
  __shared__ float lbuf[2 * TILE_FLOATS_];          // 43008 B
  const unsigned lds0 = (unsigned)(uintptr_t)&lbuf[0];   // low 32 bits = LDS byte offset
  const unsigned long long inb  = (unsigned long long)(uintptr_t)in;
  const unsigned long long outb = (unsigned long long)(uintptr_t)out;

  const unsigned td_big = 1u << 20;                 // huge tensor dims: never OOB-clip
  int parity = 0;

  for (int task = blockIdx.x; task < NTASK_; task += gridDim.x) {
    int q    = task & 3;                            // quarter-row index (QC_ == 4)
    int rest = task >> 2;
    int r    = rest % NR_;  rest /= NR_;
    int t    = rest % T_;
    int b    = rest / T_;

    const unsigned ldsa = lds0 + (unsigned)parity * (unsigned)(TILE_FLOATS_ * 4);

    // ---- load descriptor: gather one quarter patch-row into LDS ----
    unsigned long long ga = inb + 4ull *
        ((unsigned long long)(b * T_ + t) * (unsigned long long)(C_ * HW_)
         + (unsigned long long)(r * PS_) * (unsigned long long)W_
         + (unsigned long long)(q * QP_ * PS_));

    v4u g0 = { 1u,                                  // count=1, user descriptor
               ldsa,
               (unsigned)ga,
               ((unsigned)(ga >> 32) & 0x01FFFFFFu) | 0x80000000u };  // type=2 ("image")

    unsigned long long s0 = (unsigned long long)W_;   // dim0 stride (elements)
    unsigned long long s1 = (unsigned long long)HW_;  // dim1 stride (elements)
    v8i g1 = { (int)((2u << 16) | (1u << 19)),        // data_size=4B, iterate_enable=1
               (int)((td_big & 0xFFFFu) << 16),                       // tensor_dim0 lo
               (int)((td_big >> 16) | ((td_big & 0xFFFFu) << 16)),    // td0 hi | td1 lo
               (int)((td_big >> 16) | ((unsigned)PS_ << 16)),         // td1 hi | tile_dim0=16
               (int)((unsigned)PS_ | (3u << 16)),                     // tile_dim1=16, tile_dim2=3
               (int)(unsigned)s0,                                     // stride0[31:0]
               (int)((unsigned)(s0 >> 32) | (((unsigned)s1 & 0xFFFFu) << 16)),
               (int)(unsigned)(s1 >> 16) };                           // stride1[47:16]

    v4i g2l = { (int)td_big,                         // tensor_dim2 (huge)
                (int)FEAT_,                          // lds_addr_increment = 768 elems
                (int)PS_,                            // global_addr_increment = 16 elems
                (int)((unsigned)(QP_ - 1) << 16) };  // iterate_count = 6 (7 iterations)
    v4i g3  = { 0, 0, (int)(td_big >> 16), 0 };      // stride3=0, tensor_dim4 huge, tile_dim4=0

    tdm_load(g0, g1, g2l, g3);
    tdm_wait0();   // waits this load AND previous store (in-order) -> buffer swap safe

    // ---- store descriptor: 1D contiguous 5376 floats -> global ----
    unsigned long long oa = outb + 4ull *
        (((unsigned long long)b * TP_ + (unsigned long long)(t * PPF_ + r * NC_ + q * QP_))
         * (unsigned long long)FEAT_);

    v4u h0 = { 1u, ldsa, (unsigned)oa,
               ((unsigned)(oa >> 32) & 0x01FFFFFFu) | 0x80000000u };
    v8i h1 = { (int)(2u << 16),                      // data_size=4B, no iterate
               (int)((td_big & 0xFFFFu) << 16),
               (int)((td_big >> 16) | ((td_big & 0xFFFFu) << 16)),
               (int)((td_big >> 16) | ((unsigned)TILE_FLOATS_ << 16)),  // tile_dim0=5376
               0,                                    // tile_dim1=0, tile_dim2=0 (1D)
               (int)TILE_FLOATS_,                    // stride0 (unused for 1D)
               0, 0 };
    v4i h2 = { (int)td_big, (int)td_big, 0, 0 };

    tdm_store(h0, h1, h2, g3);
    parity ^= 1;
  }
  // s_endpgm does an implicit wait-idle for outstanding TENSORcnt.
#else
  // Fallback (builtin unavailable): direct float4 gather/scatter, same task split.
  for (int task = blockIdx.x; task < NTASK_; task += gridDim.x) {
    int q = task & 3;  int rest = task >> 2;
    int r = rest % NR_; rest /= NR_;
    int t = rest % T_;  int b = rest / T_;
    for (int pp = 0; pp < QP_; ++pp) {
      int c = q * QP_ + pp;
      size_t obase = ((size_t)b * TP_ + (size_t)(t * PPF_ + r * NC_ + c)) * FEAT_;
      for (int u = threadIdx.x; u < FEAT_ / 4; u += 32) {
        int ch = u >> 6; int rem = u & 63; int py = rem >> 2; int px4 = (rem & 3) << 2;
        const float4 v = *(const float4*)(in
            + (size_t)((b * T_ + t) * C_ + ch) * HW_
            + (size_t)(r * PS_ + py) * W_ + (size_t)(c * PS_ + px4));
        *(float4*)(out + obase + (size_t)u * 4) = v;
      }
    }
  }
#endif
}

// positions (t, r, c) + mask(=1). Positions may be packed as int64 (2 slots) or
// cast to float (1 slot) by the harness; decided from out_size at launch.
__global__ void moonvit_posmask(float* __restrict__ out, int posAsI64) {
  int i = blockIdx.x * blockDim.x + threadIdx.x;
  if (i >= BTP_) return;
  int j     = i % TP_;
  int frame = j / PPF_;
  int pin   = j % PPF_;
  int row   = pin / NC_;
  int col   = pin % NC_;
  if (posAsI64) {
    long long* pos = (long long*)(out + PATCH_ELEMS);
    size_t o = (size_t)i * 3;
    pos[o + 0] = frame; pos[o + 1] = row; pos[o + 2] = col;
    float* mask = out + PATCH_ELEMS + (size_t)BTP_ * 3 * 2;   // int64 = 2 float slots
    mask[i] = 1.0f;
  } else {
    float* pos = out + PATCH_ELEMS;
    size_t o = (size_t)i * 3;
    pos[o + 0] = (float)frame; pos[o + 1] = (float)row; pos[o + 2] = (float)col;
    float* mask = out + PATCH_ELEMS + (size_t)BTP_ * 3;
    mask[i] = 1.0f;
  }
}

extern "C" void kernel_launch(void* const* d_in, const int* in_sizes, int n_in,
                              void* d_out, int out_size, void* d_ws, size_t ws_size,
                              hipStream_t stream) {
  const float* in = (const float*)d_in[0];
  float* out = (float*)d_out;
  (void)in_sizes; (void)n_in; (void)d_ws; (void)ws_size;

  // float-cast positions total = PATCH + 4*BTP; int64 positions total = PATCH + 7*BTP
  int posAsI64 = (out_size > (int)(PATCH_ELEMS + 4ull * BTP_)) ? 1 : 0;

  // 2048 single-wave workgroups, each grid-striding over 14336 TDM tasks
  // (7 tasks each) with double-buffered LDS so loads overlap stores.
  moonvit_patch_pack<<<2048, 32, 0, stream>>>(in, out);
  moonvit_posmask<<<(BTP_ + 255) / 256, 256, 0, stream>>>(out, posAsI64);
}